// GraphConv_38594576122649
// MI455X (gfx1250) — compile-verified
//
#include <hip/hip_runtime.h>
#include <hip/hip_bf16.h>
#include <cstddef>

// ---- problem constants (match reference) ----
#define Mn     4096     // nodes
#define NN     16       // neighbors per node
#define Gn     4        // graphs
#define IN_F   32
#define OUT_F  32
#define LOC_F  8
#define LHID   128
#define Bn     16
#define PAIRS  (Mn * NN)       // 65536 pairs per graph

typedef __attribute__((ext_vector_type(2))) float v2f;
typedef __attribute__((ext_vector_type(8))) float v8f;

// Branchless tanh: tanh(x) = 1 - 2/(exp(2x)+1).  v_exp_f32 + raw v_rcp_f32
// (__builtin_amdgcn_rcpf, ~1ulp) -- NOT __frcp_rn, which lowers to the full
// IEEE div_scale/div_fmas/div_fixup sequence.  Clamp keeps exp finite.
__device__ __forceinline__ float fast_tanh(float x) {
    x = fminf(fmaxf(x, -15.0f), 15.0f);
    float e = __expf(2.0f * x);
    return 1.0f - 2.0f * __builtin_amdgcn_rcpf(e + 1.0f);
}

// ============================================================================
// Kernel 1: init output with bias bx (out[b,i,o] = bx[o]); clears poison.
// out has B*M*OUT_F = 2,097,152 floats -> 524,288 float4 threads.
// ============================================================================
__global__ void gc_init_kernel(const float* __restrict__ bx,
                               float* __restrict__ out) {
    int tid = blockIdx.x * blockDim.x + threadIdx.x;      // 0 .. 524287
    int o = (tid & 7) * 4;                                // OUT_F=32 -> 8 float4 per row
    float4 v = make_float4(bx[o], bx[o + 1], bx[o + 2], bx[o + 3]);
    reinterpret_cast<float4*>(out)[tid] = v;
}

// ============================================================================
// Kernel 2: attention.  One wave32 per (graph, node).
//   h   = tanh(maps_tile(16x8) @ W1[g]^T + b1[g])  via V_WMMA_F32_16X16X4_F32
//   ctx = h . W2[g] + b2[g]   (per pair-row)
//   attn = softmax over the node's 16 neighbors (== the 16-row WMMA tile)
// A layout (16x4 f32): lanes 0-15 hold (K0,K1) of row (lane&15),
//                      lanes 16-31 hold (K2,K3).  B mirrored with N in lanes.
// C/D (16x16 f32): vgpr r, lanes 0-15 -> (M=r, N=lane); lanes 16-31 -> (M=r+8).
// ============================================================================
__global__ void gc_attn_kernel(const float* __restrict__ maps,
                               const float* __restrict__ W1,
                               const float* __restrict__ b1v,
                               const float* __restrict__ W2,
                               const float* __restrict__ b2v,
                               float* __restrict__ attn) {
    const int lane = threadIdx.x & 31;
    const int tile = blockIdx.x * (blockDim.x >> 5) + (threadIdx.x >> 5);
    const int g    = tile >> 12;          // 4096 node-tiles per graph
    const int node = tile & 4095;
    const int half = lane >> 4;           // 0: K/M low half, 1: high half
    const int l15  = lane & 15;
    const int koff = half * 2;

    // A: maps rows [node*16 .. node*16+15], LOC_F = 8 = two K=4 chunks
    const float* mp = maps + (size_t)(node * NN + l15) * LOC_F;
    v2f a0 = { mp[koff],     mp[koff + 1] };
    v2f a1 = { mp[4 + koff], mp[5 + koff] };

    const float b2g = b2v[g];
    v8f acc = {};   // per-lane partial of sum_h tanh(h)*W2[h]

    #pragma unroll
    for (int ht = 0; ht < LHID / 16; ++ht) {
        const int h = ht * 16 + l15;                      // hidden column for this lane
        const float* wp = W1 + ((size_t)g * LHID + h) * LOC_F;
        v2f bb0 = { wp[koff],     wp[koff + 1] };         // B[k][n] = W1[g][h][k]
        v2f bb1 = { wp[4 + koff], wp[5 + koff] };

        v8f c = {};
        c = __builtin_amdgcn_wmma_f32_16x16x4_f32(false, a0, false, bb0,
                                                  (short)0, c, false, false);
        c = __builtin_amdgcn_wmma_f32_16x16x4_f32(false, a1, false, bb1,
                                                  (short)0, c, false, false);

        const float bias = b1v[g * LHID + h];
        const float w2   = W2[g * LHID + h];
        #pragma unroll
        for (int r = 0; r < 8; ++r)
            acc[r] += fast_tanh(c[r] + bias) * w2;
    }

    // Reduce over the 16 N-lanes of each half (butterfly -> all lanes hold sum)
    #pragma unroll
    for (int r = 0; r < 8; ++r) {
        float v = acc[r];
        v += __shfl_xor(v, 1, 32);
        v += __shfl_xor(v, 2, 32);
        v += __shfl_xor(v, 4, 32);
        v += __shfl_xor(v, 8, 32);
        acc[r] = v + b2g;       // ctx for row (half*8 + r)
    }

    // Softmax over all 16 rows: 8 in low half-lanes, 8 in high half-lanes.
    float lmax = acc[0];
    #pragma unroll
    for (int r = 1; r < 8; ++r) lmax = fmaxf(lmax, acc[r]);
    float gmax = fmaxf(lmax, __shfl_xor(lmax, 16, 32));

    float e[8];
    float lsum = 0.f;
    #pragma unroll
    for (int r = 0; r < 8; ++r) { e[r] = __expf(acc[r] - gmax); lsum += e[r]; }
    float gsum = lsum + __shfl_xor(lsum, 16, 32);
    float inv  = __builtin_amdgcn_rcpf(gsum);

    if (l15 == 0) {     // lanes 0 and 16 write their 8 rows
        float* ap = attn + (size_t)g * PAIRS + node * NN + half * 8;
        #pragma unroll
        for (int r = 0; r < 8; ++r) ap[r] = e[r] * inv;
    }
}

// ============================================================================
// Kernel 3: y[g, b, j, o] = sum_f x[b,j,f] * Wx[g,f,o].
// Rows r = b*M + j (contiguous in x).  One wave per (graph, 16-row tile);
// K=32 -> 8 chunks of K=4; N=32 -> 2 N-tiles of 16.  16 f32 WMMAs per wave.
// ============================================================================
__global__ void gc_xwx_kernel(const float* __restrict__ x,
                              const float* __restrict__ Wx,
                              float* __restrict__ y) {
    const int lane = threadIdx.x & 31;
    const int tile = blockIdx.x * (blockDim.x >> 5) + (threadIdx.x >> 5);
    const int g    = tile >> 12;          // 65536/16 = 4096 row-tiles per graph
    const int rt   = tile & 4095;
    const int half = lane >> 4;
    const int l15  = lane & 15;
    const int koff = half * 2;

    const float* xr = x + (size_t)(rt * 16 + l15) * IN_F;
    v2f a[8];
    #pragma unroll
    for (int kc = 0; kc < 8; ++kc)
        a[kc] = (v2f){ xr[kc * 4 + koff], xr[kc * 4 + koff + 1] };

    const float* wg = Wx + (size_t)g * IN_F * OUT_F;

    #pragma unroll
    for (int nt = 0; nt < 2; ++nt) {
        const int ncol = nt * 16 + l15;
        v8f c = {};
        #pragma unroll
        for (int kc = 0; kc < 8; ++kc) {
            v2f bb = { wg[(kc * 4 + koff) * OUT_F + ncol],
                       wg[(kc * 4 + koff + 1) * OUT_F + ncol] };
            c = __builtin_amdgcn_wmma_f32_16x16x4_f32(false, a[kc], false, bb,
                                                      (short)0, c, false, false);
        }
        #pragma unroll
        for (int r = 0; r < 8; ++r) {
            const int m = half * 8 + r;
            y[((size_t)g * (Bn * Mn) + rt * 16 + m) * OUT_F + ncol] = c[r];
        }
    }
}

// ============================================================================
// Kernel 4: sparse scatter.  thread <-> (g, pair, b).
//   row = L_idx[n] / M, col = L_idx[n] % M  (M = 4096, power of 2)
//   out[b, row, :] += attn[g,n] * y[g, b, col, :]
// ============================================================================
__global__ void gc_scatter_kernel(const int* __restrict__ L_idx,
                                  const float* __restrict__ attn,
                                  const float* __restrict__ y,
                                  float* __restrict__ out) {
    const int tid = blockIdx.x * blockDim.x + threadIdx.x;
    const int b   = tid & (Bn - 1);
    const int pn  = tid >> 4;             // g*PAIRS + n
    const int g   = pn >> 16;             // PAIRS = 65536
    const int n   = pn & (PAIRS - 1);

    const int idx = L_idx[n];
    const int row = idx >> 12;            // / 4096
    const int col = idx & (Mn - 1);
    const float val = attn[(size_t)g * PAIRS + n];

    const float4* yv =
        reinterpret_cast<const float4*>(y + ((size_t)g * (Bn * Mn) +
                                             (size_t)b * Mn + col) * OUT_F);
    float* op = out + ((size_t)b * Mn + row) * OUT_F;

    #pragma unroll
    for (int q = 0; q < OUT_F / 4; ++q) {
        float4 v = yv[q];
        atomicAdd(op + q * 4 + 0, val * v.x);
        atomicAdd(op + q * 4 + 1, val * v.y);
        atomicAdd(op + q * 4 + 2, val * v.z);
        atomicAdd(op + q * 4 + 3, val * v.w);
    }
}

// ============================================================================
extern "C" void kernel_launch(void* const* d_in, const int* in_sizes, int n_in,
                              void* d_out, int out_size, void* d_ws, size_t ws_size,
                              hipStream_t stream) {
    const float* x    = (const float*)d_in[0];   // (B, M, IN_F)
    const float* maps = (const float*)d_in[1];   // (M*NN, LOC_F)
    const int*   Lidx = (const int*)  d_in[2];   // (M*NN,)
    const float* W1   = (const float*)d_in[3];   // (G, LHID, LOC_F)
    const float* b1   = (const float*)d_in[4];   // (G, LHID)
    const float* W2   = (const float*)d_in[5];   // (G, LHID)
    const float* b2   = (const float*)d_in[6];   // (G,)
    const float* Wx   = (const float*)d_in[7];   // (G, IN_F, OUT_F)
    const float* bx   = (const float*)d_in[8];   // (OUT_F,)
    float* out = (float*)d_out;                  // (B, M, OUT_F)

    // workspace layout: attn (G*PAIRS floats = 1 MB) | y (G*B*M*OUT_F = 33.5 MB)
    float* attn = (float*)d_ws;
    float* y    = attn + (size_t)Gn * PAIRS;

    // 1) out <- bx broadcast
    {
        int threads = (Bn * Mn * OUT_F) / 4;     // 524288
        gc_init_kernel<<<threads / 256, 256, 0, stream>>>(bx, out);
    }
    // 2) attention MLP + softmax (G*M = 16384 waves, 4 waves/block)
    {
        int tiles = Gn * Mn;
        gc_attn_kernel<<<tiles / 4, 128, 0, stream>>>(maps, W1, b1, W2, b2, attn);
    }
    // 3) y = x @ Wx per graph (G * (B*M/16) = 16384 waves)
    {
        int tiles = Gn * (Bn * Mn / 16);
        gc_xwx_kernel<<<tiles / 4, 128, 0, stream>>>(x, Wx, y);
    }
    // 4) sparse scatter-accumulate into out
    {
        int threads = Gn * PAIRS * Bn;           // 4,194,304
        gc_scatter_kernel<<<threads / 256, 256, 0, stream>>>(Lidx, attn, y, out);
    }
}